// OnlineCosineTripletLoss_11123965296712
// MI455X (gfx1250) — compile-verified
//
#include <hip/hip_runtime.h>
#include <hip/hip_bf16.h>

#define MARGIN 0.1f

typedef __attribute__((ext_vector_type(2))) float v2f;
typedef __attribute__((ext_vector_type(8))) float v8f;

#define N 512
#define D 128

// ---------------------------------------------------------------------------
// K1: row L2-normalize: e[row] = emb[row] / ||emb[row]||
// 512 blocks x 128 threads (one thread per element of the row)
// ---------------------------------------------------------------------------
__global__ void normalize_kernel(const float* __restrict__ emb,
                                 float* __restrict__ e) {
    const int row = blockIdx.x;
    const int t   = threadIdx.x;              // 0..127
    float v = emb[row * D + t];

    __shared__ float red[D];
    red[t] = v * v;
    __syncthreads();
#pragma unroll
    for (int s = D / 2; s > 0; s >>= 1) {
        if (t < s) red[t] += red[t + s];
        __syncthreads();
    }
    const float inv = 1.0f / sqrtf(red[0]);
    e[row * D + t] = v * inv;
}

// ---------------------------------------------------------------------------
// K2: S = e * e^T via V_WMMA_F32_16X16X4_F32 (full fp32 matrix pipe).
// One wave32 per 16x16 output tile; K stepped by 4 (32 WMMAs / tile).
//
// A (16x4 f32): lane m(0..15) holds A[m][K=0..1] in 2 VGPRs, lanes 16..31
// hold K=2..3.  B (4x16) mirrors with lane = column n.  Since B = e^T tile,
// B[k][n] = e[colBase+n][k]  ->  identical b64 load pattern as A.
// ---------------------------------------------------------------------------
__global__ void gemm_wmma_kernel(const float* __restrict__ e,
                                 float* __restrict__ S) {
    const int lane  = threadIdx.x;            // 0..31
    const int tileM = blockIdx.x * 16;
    const int tileN = blockIdx.y * 16;

    const int mn   = lane & 15;               // row (A) / col (B) within tile
    const int koff = (lane >> 4) * 2;         // 0 for lanes 0-15, 2 for 16-31

    const float* arow = e + (tileM + mn) * D + koff;
    const float* brow = e + (tileN + mn) * D + koff;

    v8f acc = {};
#pragma unroll
    for (int k = 0; k < D; k += 4) {
        v2f a = *(const v2f*)(arow + k);
        v2f b = *(const v2f*)(brow + k);
        acc = __builtin_amdgcn_wmma_f32_16x16x4_f32(
            /*neg_a=*/false, a, /*neg_b=*/false, b,
            /*c_mod=*/(short)0, acc, /*reuse_a=*/false, /*reuse_b=*/false);
    }

    // C/D layout: lanes 0-15 -> rows tileM+0..7, lanes 16-31 -> rows tileM+8..15
    const int col     = tileN + mn;
    const int rowBase = tileM + (lane >> 4) * 8;
#pragma unroll
    for (int r = 0; r < 8; ++r) {
        S[(rowBase + r) * N + col] = acc[r];
    }
}

// ---------------------------------------------------------------------------
// K3: per-anchor triplet pass. Mask collapses to:
//   p != a, labels[p]==labels[a], labels[n]!=labels[a]
// One block per anchor; S[a,:] and labels staged in LDS.
// Deterministic block tree reduction -> partial[a] = sum, partial[N+a] = count
// ---------------------------------------------------------------------------
__global__ void triplet_kernel(const float* __restrict__ S,
                               const int* __restrict__ labels,
                               float* __restrict__ partial) {
    const int a  = blockIdx.x;
    const int t  = threadIdx.x;               // 0..255
    __shared__ float sS[N];
    __shared__ int   sl[N];

    for (int i = t; i < N; i += 256) {
        sS[i] = S[a * N + i];
        sl[i] = labels[i];
    }
    __syncthreads();

    const int la = sl[a];
    float sum = 0.0f, cnt = 0.0f;

    for (int p = 0; p < N; ++p) {
        if (p == a || sl[p] != la) continue;
        const float base = MARGIN - sS[p];    // loss = relu(S[a,n] + base)
        for (int n = t; n < N; n += 256) {
            if (sl[n] == la) continue;        // implies n!=a, n!=p
            float l = sS[n] + base;
            l = fmaxf(l, 0.0f);
            sum += l;
            if (l > 1e-16f) cnt += 1.0f;
        }
    }

    __shared__ float rs[256];
    __shared__ float rc[256];
    rs[t] = sum;
    rc[t] = cnt;
    __syncthreads();
#pragma unroll
    for (int s = 128; s > 0; s >>= 1) {
        if (t < s) { rs[t] += rs[t + s]; rc[t] += rc[t + s]; }
        __syncthreads();
    }
    if (t == 0) {
        partial[a]     = rs[0];
        partial[N + a] = rc[0];
    }
}

// ---------------------------------------------------------------------------
// K4: final deterministic reduce of 512 per-anchor partials -> scalar
// ---------------------------------------------------------------------------
__global__ void finalize_kernel(const float* __restrict__ partial,
                                float* __restrict__ out) {
    const int t = threadIdx.x;                // 0..511
    __shared__ float rs[N];
    __shared__ float rc[N];
    rs[t] = partial[t];
    rc[t] = partial[N + t];
    __syncthreads();
#pragma unroll
    for (int s = N / 2; s > 0; s >>= 1) {
        if (t < s) { rs[t] += rs[t + s]; rc[t] += rc[t + s]; }
        __syncthreads();
    }
    if (t == 0) out[0] = rs[0] / (rc[0] + 1e-16f);
}

// ---------------------------------------------------------------------------
extern "C" void kernel_launch(void* const* d_in, const int* in_sizes, int n_in,
                              void* d_out, int out_size, void* d_ws, size_t ws_size,
                              hipStream_t stream) {
    const float* emb    = (const float*)d_in[0];  // (512,128) fp32
    const int*   labels = (const int*)d_in[1];    // (512,) int32
    float*       out    = (float*)d_out;          // scalar fp32

    float* ws      = (float*)d_ws;
    float* e       = ws;                 // 512*128 floats (256 KB)
    float* S       = ws + N * D;         // 512*512 floats (1 MB)
    float* partial = S + N * N;          // 1024 floats

    normalize_kernel<<<N, D, 0, stream>>>(emb, e);

    dim3 gg(N / 16, N / 16);             // 32x32 tiles, one wave32 each
    gemm_wmma_kernel<<<gg, 32, 0, stream>>>(e, S);

    triplet_kernel<<<N, 256, 0, stream>>>(S, labels, partial);

    finalize_kernel<<<1, N, 0, stream>>>(partial, out);
}